// encoder_48679159333591
// MI455X (gfx1250) — compile-verified
//
#include <hip/hip_runtime.h>
#include <hip/hip_bf16.h>

typedef __attribute__((ext_vector_type(2))) float v2f;
typedef __attribute__((ext_vector_type(8))) float v8f;

#define BM 64
#define BN 128
#define BK 32
#define A_STRIDE 36     // floats per A row (padded): rows map to distinct banks, half-waves disjoint parity
#define BP_STRIDE 144   // float2 per B pair-row (padded): pair-row step = 288 dwords = +32 banks

// -------------------- GEMM: C[M,N] = A[M,K] * B[K,N]  (fp32 WMMA) --------------------
// Block tile 64x128, 8 waves in a 2x4 grid, each wave owns a 32x32 output (4 accumulators).
// B is staged in LDS packed as K-pairs so each WMMA B fragment is one contiguous ds_load_b64.
__global__ __launch_bounds__(256) void gcn_gemm_f32(
    const float* __restrict__ A, const float* __restrict__ Bw,
    float* __restrict__ C, int M, int K, int N)
{
    __shared__ float  As[BM * A_STRIDE];            // 9216 B
    __shared__ float2 Bp[(BK / 2) * BP_STRIDE];     // 18432 B

    const int tid  = threadIdx.x;
    const int lane = tid & 31;
    const int wave = tid >> 5;     // 0..7
    const int mt   = wave >> 2;    // 0..1 : 32-row slab
    const int nt   = wave & 3;     // 0..3 : 32-col slab
    const int l15  = lane & 15;
    const int lhi  = lane >> 4;    // 0/1 half-wave

    const int block_m = blockIdx.x * BM;
    const int block_n = blockIdx.y * BN;

    v8f acc00 = {}, acc01 = {}, acc10 = {}, acc11 = {};

    for (int k0 = 0; k0 < K; k0 += BK) {
        // ---- A tile 64x32: two float4 per thread ----
        #pragma unroll
        for (int rr = 0; rr < BM; rr += 32) {
            int r  = rr + (tid >> 3);    // 0..63
            int c  = (tid & 7) * 4;      // 0..28
            int gr = block_m + r;
            float4 v = make_float4(0.f, 0.f, 0.f, 0.f);
            if (gr < M) v = *(const float4*)(A + (size_t)gr * K + k0 + c);
            *(float4*)(&As[r * A_STRIDE + c]) = v;
        }
        // ---- B tile 32x128, packed as K-pairs: Bp[p][n] = {B[2p][n], B[2p+1][n]} ----
        #pragma unroll
        for (int pp = 0; pp < BK / 2; pp += 8) {
            int p = pp + (tid >> 5);     // pair-row 0..15
            int c = (tid & 31) * 4;      // col 0..124
            const float* brow0 = Bw + (size_t)(k0 + 2 * p) * N + block_n + c;
            float4 r0 = *(const float4*)(brow0);
            float4 r1 = *(const float4*)(brow0 + N);
            float2* dst = &Bp[p * BP_STRIDE + c];
            dst[0] = make_float2(r0.x, r1.x);
            dst[1] = make_float2(r0.y, r1.y);
            dst[2] = make_float2(r0.z, r1.z);
            dst[3] = make_float2(r0.w, r1.w);
        }
        __syncthreads();

        #pragma unroll
        for (int kk = 0; kk < BK; kk += 4) {
            // A fragments: lanes 0-15 -> K=kk,kk+1 ; lanes 16-31 -> K=kk+2,kk+3
            int ak  = kk + 2 * lhi;                       // even -> 8B aligned
            int ar0 = (mt * 32 + l15) * A_STRIDE + ak;
            v2f a0 = *(const v2f*)(&As[ar0]);
            v2f a1 = *(const v2f*)(&As[ar0 + 16 * A_STRIDE]);
            // B fragments: one contiguous pair per lane
            int p = (kk >> 1) + lhi;
            const float2* bptr = &Bp[p * BP_STRIDE + nt * 32 + l15];
            v2f b0 = *(const v2f*)(bptr);
            v2f b1 = *(const v2f*)(bptr + 16);

            acc00 = __builtin_amdgcn_wmma_f32_16x16x4_f32(false, a0, false, b0,
                                                          (short)0, acc00, false, false);
            acc01 = __builtin_amdgcn_wmma_f32_16x16x4_f32(false, a0, false, b1,
                                                          (short)0, acc01, false, false);
            acc10 = __builtin_amdgcn_wmma_f32_16x16x4_f32(false, a1, false, b0,
                                                          (short)0, acc10, false, false);
            acc11 = __builtin_amdgcn_wmma_f32_16x16x4_f32(false, a1, false, b1,
                                                          (short)0, acc11, false, false);
        }
        __syncthreads();
    }

    // ---- store D tiles: VGPR r -> row (r + 8*lhi), col = l15 within each 16-tile ----
    int n0 = block_n + nt * 32 + l15;
    #pragma unroll
    for (int r = 0; r < 8; ++r) {
        int gm0 = block_m + mt * 32 + r + 8 * lhi;
        int gm1 = gm0 + 16;
        if (gm0 < M) {
            C[(size_t)gm0 * N + n0]      = acc00[r];
            C[(size_t)gm0 * N + n0 + 16] = acc01[r];
        }
        if (gm1 < M) {
            C[(size_t)gm1 * N + n0]      = acc10[r];
            C[(size_t)gm1 * N + n0 + 16] = acc11[r];
        }
    }
}

// -------------------- degree with self loop --------------------
__global__ void deg_init(float* __restrict__ deg, int n) {
    int i = blockIdx.x * blockDim.x + threadIdx.x;
    if (i < n) deg[i] = 1.0f;   // self-loop
}

__global__ void deg_scatter(const long long* __restrict__ ei, float* __restrict__ deg, int E) {
    int e = blockIdx.x * blockDim.x + threadIdx.x;
    if (e < E) atomicAdd(&deg[(int)ei[(size_t)E + e]], 1.0f);  // in-degree over dst
}

__global__ void deg_finalize(float* __restrict__ dinv, int n) {
    int i = blockIdx.x * blockDim.x + threadIdx.x;
    if (i < n) dinv[i] = rsqrtf(dinv[i]);
}

// -------------------- self-loop init: agg = h * dinv^2 --------------------
__global__ void selfloop_init(const float* __restrict__ h, const float* __restrict__ dinv,
                              float* __restrict__ agg, int n, int C) {
    size_t idx = (size_t)blockIdx.x * blockDim.x + threadIdx.x;
    size_t total = (size_t)n * C;
    if (idx < total) {
        int node = (int)(idx / C);
        float di = dinv[node];
        agg[idx] = h[idx] * di * di;
    }
}

// -------------------- edge scatter: agg[dst] += coef * h[src] --------------------
__global__ __launch_bounds__(256) void scatter_edges(
    const float* __restrict__ h, const long long* __restrict__ ei,
    const float* __restrict__ dinv, float* __restrict__ agg, int E, int C)
{
    int e = blockIdx.x;
    int s = (int)ei[e];
    int d = (int)ei[(size_t)E + e];
    float coef = dinv[s] * dinv[d];
    const float* hs = h + (size_t)s * C;
    float* ad = agg + (size_t)d * C;
    for (int c = threadIdx.x; c < C; c += 256)
        atomicAdd(&ad[c], hs[c] * coef);
}

// -------------------- bias + ReLU + LayerNorm (one block per node) --------------------
template <int C>
__global__ __launch_bounds__(256) void bias_relu_ln(
    const float* __restrict__ agg, const float* __restrict__ bias,
    const float* __restrict__ gamma, const float* __restrict__ beta,
    float* __restrict__ out)
{
    constexpr int PER = C / 256;
    __shared__ float red[256];
    const int node = blockIdx.x;
    const float* row = agg + (size_t)node * C;

    float v[PER];
    float s = 0.f;
    #pragma unroll
    for (int i = 0; i < PER; ++i) {
        int c = threadIdx.x + i * 256;
        float t = row[c] + bias[c];
        t = t > 0.f ? t : 0.f;          // ReLU
        v[i] = t;
        s += t;
    }
    red[threadIdx.x] = s;
    __syncthreads();
    for (int o = 128; o > 0; o >>= 1) {
        if (threadIdx.x < o) red[threadIdx.x] += red[threadIdx.x + o];
        __syncthreads();
    }
    float mean = red[0] * (1.0f / C);
    __syncthreads();

    float s2 = 0.f;
    #pragma unroll
    for (int i = 0; i < PER; ++i) {
        float d = v[i] - mean;
        s2 += d * d;
    }
    red[threadIdx.x] = s2;
    __syncthreads();
    for (int o = 128; o > 0; o >>= 1) {
        if (threadIdx.x < o) red[threadIdx.x] += red[threadIdx.x + o];
        __syncthreads();
    }
    float rinv = rsqrtf(red[0] * (1.0f / C) + 1e-5f);

    float* orow = out + (size_t)node * C;
    #pragma unroll
    for (int i = 0; i < PER; ++i) {
        int c = threadIdx.x + i * 256;
        orow[c] = (v[i] - mean) * rinv * gamma[c] + beta[c];
    }
}

// -------------------- host orchestration --------------------
extern "C" void kernel_launch(void* const* d_in, const int* in_sizes, int n_in,
                              void* d_out, int out_size, void* d_ws, size_t ws_size,
                              hipStream_t stream) {
    const float*      x  = (const float*)d_in[0];
    const long long*  ei = (const long long*)d_in[1];
    const float* W1 = (const float*)d_in[2];
    const float* b1 = (const float*)d_in[3];
    const float* g1 = (const float*)d_in[4];
    const float* be1 = (const float*)d_in[5];
    const float* W2 = (const float*)d_in[6];
    const float* b2 = (const float*)d_in[7];
    const float* g2 = (const float*)d_in[8];
    const float* be2 = (const float*)d_in[9];
    const float* W3 = (const float*)d_in[10];
    const float* b3 = (const float*)d_in[11];
    const float* g3 = (const float*)d_in[12];
    const float* be3 = (const float*)d_in[13];

    const int N = in_sizes[0] / 256;  // 50000 nodes
    const int E = in_sizes[1] / 2;    // 500000 edges

    // workspace layout (floats)
    float* ws   = (float*)d_ws;
    float* dinv = ws;
    size_t off  = ((size_t)N + 255) & ~(size_t)255;
    float* hbuf = dinv + off;                         // up to N x 1024
    float* aggb = hbuf + (size_t)N * 1024;            // up to N x 1024
    float* xbuf = aggb + (size_t)N * 1024;            // up to N x 1024
    (void)ws_size;

    // ---- degrees (with self loops) -> dinv ----
    deg_init<<<(N + 255) / 256, 256, 0, stream>>>(dinv, N);
    deg_scatter<<<(E + 255) / 256, 256, 0, stream>>>(ei, dinv, E);
    deg_finalize<<<(N + 255) / 256, 256, 0, stream>>>(dinv, N);

    const int gm = (N + BM - 1) / BM;

    // ================= Layer 1: 256 -> 1024 =================
    {
        const int K = 256, C = 1024;
        dim3 grid(gm, C / BN);
        gcn_gemm_f32<<<grid, 256, 0, stream>>>(x, W1, hbuf, N, K, C);
        size_t total = (size_t)N * C;
        selfloop_init<<<(int)((total + 255) / 256), 256, 0, stream>>>(hbuf, dinv, aggb, N, C);
        scatter_edges<<<E, 256, 0, stream>>>(hbuf, ei, dinv, aggb, E, C);
        bias_relu_ln<1024><<<N, 256, 0, stream>>>(aggb, b1, g1, be1, xbuf);
    }
    // ================= Layer 2: 1024 -> 512 =================
    {
        const int K = 1024, C = 512;
        dim3 grid(gm, C / BN);
        gcn_gemm_f32<<<grid, 256, 0, stream>>>(xbuf, W2, hbuf, N, K, C);
        size_t total = (size_t)N * C;
        selfloop_init<<<(int)((total + 255) / 256), 256, 0, stream>>>(hbuf, dinv, aggb, N, C);
        scatter_edges<<<E, 256, 0, stream>>>(hbuf, ei, dinv, aggb, E, C);
        bias_relu_ln<512><<<N, 256, 0, stream>>>(aggb, b2, g2, be2, xbuf);
    }
    // ================= Layer 3: 512 -> 256 =================
    {
        const int K = 512, C = 256;
        dim3 grid(gm, C / BN);
        gcn_gemm_f32<<<grid, 256, 0, stream>>>(xbuf, W3, hbuf, N, K, C);
        size_t total = (size_t)N * C;
        selfloop_init<<<(int)((total + 255) / 256), 256, 0, stream>>>(hbuf, dinv, aggb, N, C);
        scatter_edges<<<E, 256, 0, stream>>>(hbuf, ei, dinv, aggb, E, C);
        bias_relu_ln<256><<<N, 256, 0, stream>>>(aggb, b3, g3, be3, (float*)d_out);
    }
}